// KNNSmoothingLoss_46557445488920
// MI455X (gfx1250) — compile-verified
//
#include <hip/hip_runtime.h>
#include <hip/hip_bf16.h>
#include <math.h>

// ---------------------------------------------------------------------------
// KNN smoothing loss for MI455X (gfx1250, wave32).
//
// dist[m][n] = sq_m + sq_n - 2 x_m.x_n computed 16x16 tile at a time with
// V_WMMA_F32_16X16X4_F32:
//   A row m (candidate, loop) = (-2x_m, -2y_m, -2z_m, sq_m)  (16x4)
//   B col n (query, fixed)    = ( x_n,   y_n,   z_n,  1   )  (4x16)
//   C[m][n] = sq_n broadcast -> per-lane constant, loop-invariant
//
// One wave owns 16 QUERY points (the B columns); lane L holds 8 candidate
// distances for query q0+(L%16) directly in the WMMA D registers. No LDS,
// no barriers in the hot loop.
//
// Top-17 lists are uint32 bit patterns (clamped >= 0, so IEEE order ==
// unsigned order) -> v_min_u32/v_max_u32 insertion network, no NaN
// canonicalization. The insert guard is per-TILE: min-of-8 vs s[16], so the
// guarded block (and its unavoidable phi register copies) fires per tile,
// not per element; inside, all 8 inserts run branchless.
// ---------------------------------------------------------------------------

#define BATCH 16
#define NPTS  4096
#define KNN   16
#define NTILE (NPTS / 16)
#define ALPHA 1.05f

typedef __attribute__((ext_vector_type(2))) float v2f;
typedef __attribute__((ext_vector_type(8))) float v8f;

__device__ __forceinline__ unsigned int umin2(unsigned int a, unsigned int b) {
    return a < b ? a : b;
}

__device__ __forceinline__ void topk_insert17_u32(unsigned int (&s)[17],
                                                  unsigned int v) {
    // s sorted ascending (as uint == float order); push v through,
    // old max falls off the end.
    #pragma unroll
    for (int q = 0; q < 17; ++q) {
        const unsigned int lo = (s[q] < v) ? s[q] : v;
        const unsigned int hi = (s[q] < v) ? v : s[q];
        s[q] = lo;
        v = hi;
    }
}

__global__ __launch_bounds__(32) void knn_tile_kernel(
    const float* __restrict__ pcs, float* __restrict__ knn_out) {
    const int wg   = blockIdx.x;          // 0 .. BATCH*NTILE-1
    const int b    = wg >> 8;             // NTILE == 256
    const int q0   = (wg & 255) << 4;     // base of this wave's 16 queries
    const int lane = threadIdx.x;         // 0..31 (wave32)
    const int m    = lane & 15;
    const int hi   = lane >> 4;

    const float* cloud = pcs + (size_t)b * NPTS * 3;

    // ---- B matrix + C broadcast: the 16 fixed query points ----
    const float* qp = cloud + (size_t)(q0 + m) * 3;
    const float qx = qp[0], qy = qp[1], qz = qp[2];
    const float qsq = qx * qx + qy * qy + qz * qz;
    v2f bm;
    bm.x = hi ? qz   : qx;                // K2 : K0
    bm.y = hi ? 1.0f : qy;                // K3 : K1
    v8f cinit;
    #pragma unroll
    for (int k = 0; k < 8; ++k) cinit[k] = qsq;   // C[m][n] = sq_query(n)

    // ---- per-(query, row-half) running top-17 smallest, sorted ascending,
    //      stored as uint32 bit patterns (valid: all values >= 0) ----
    unsigned int s[17];
    const unsigned int SENT = __float_as_uint(3.0e38f);
    #pragma unroll
    for (int k = 0; k < 17; ++k) s[k] = SENT;

    for (int t = 0; t < NTILE; ++t) {
        // A matrix: candidate tile t. Lane supplies row M = lane%16.
        const float* cp = cloud + (size_t)((t << 4) + m) * 3;
        const float cx = cp[0], cy = cp[1], cz = cp[2];
        __builtin_prefetch(cp + 48, 0, 1);        // next tile, same lane
        const float csq = cx * cx + cy * cy + cz * cz;
        v2f a;
        a.x = hi ? (-2.0f * cz) : (-2.0f * cx);   // K2 : K0
        a.y = hi ? csq          : (-2.0f * cy);   // K3 : K1

        // D = A*B + C -> lane holds 8 distances for its query, rows r+8*hi.
        v8f d = __builtin_amdgcn_wmma_f32_16x16x4_f32(
            false, a, false, bm, (short)0, cinit, false, false);

        // Clamp to [0, 3e38] (one v_med3_f32 each) and reinterpret as u32.
        unsigned int dv[8];
        #pragma unroll
        for (int k = 0; k < 8; ++k)
            dv[k] = __float_as_uint(
                __builtin_amdgcn_fmed3f(d[k], 0.0f, 3.0e38f));

        // Tile-level guard: skip all 8 inserts unless this lane's best
        // candidate beats the current 17th smallest.
        const unsigned int mn =
            umin2(umin2(umin2(dv[0], dv[1]), umin2(dv[2], dv[3])),
                  umin2(umin2(dv[4], dv[5]), umin2(dv[6], dv[7])));
        if (mn < s[16]) {
            #pragma unroll
            for (int k = 0; k < 8; ++k) topk_insert17_u32(s, dv[k]);
        }
    }

    // Merge the two row-half top-17 lists (partner = lane ^ 16).
    #pragma unroll
    for (int k = 0; k < 17; ++k) {
        const unsigned int v =
            (unsigned int)__shfl((int)s[k], lane ^ 16, 32);
        if (hi == 0 && v < s[16]) topk_insert17_u32(s, v);
    }

    if (hi == 0) {
        float sum = 0.0f;
        #pragma unroll
        for (int k = 0; k < 17; ++k) sum += __uint_as_float(s[k]);
        // drop the smallest (self distance ~0), mean over K=16 neighbors
        const float knn = (sum - __uint_as_float(s[0])) * (1.0f / (float)KNN);
        knn_out[(size_t)b * NPTS + q0 + m] = knn;
    }
}

__global__ __launch_bounds__(256) void batch_stats_kernel(
    const float* __restrict__ knn, float* __restrict__ psums) {
    __shared__ float red[256];
    __shared__ float red2[256];
    const int b = blockIdx.x;
    const float* kb = knn + (size_t)b * NPTS;

    float sum = 0.0f, sumsq = 0.0f;
    for (int i = threadIdx.x; i < NPTS; i += 256) {
        const float v = kb[i];
        sum += v;
        sumsq += v * v;
    }
    red[threadIdx.x] = sum;
    red2[threadIdx.x] = sumsq;
    __syncthreads();
    for (int off = 128; off > 0; off >>= 1) {
        if (threadIdx.x < off) {
            red[threadIdx.x]  += red[threadIdx.x + off];
            red2[threadIdx.x] += red2[threadIdx.x + off];
        }
        __syncthreads();
    }
    const float mu  = red[0] * (1.0f / NPTS);
    const float var = fmaxf(red2[0] * (1.0f / NPTS) - mu * mu, 0.0f);
    const float thresh = mu + ALPHA * sqrtf(var);
    __syncthreads();

    float p = 0.0f;
    for (int i = threadIdx.x; i < NPTS; i += 256) {
        const float v = kb[i];
        p += (v > thresh) ? v : 0.0f;
    }
    red[threadIdx.x] = p;
    __syncthreads();
    for (int off = 128; off > 0; off >>= 1) {
        if (threadIdx.x < off) red[threadIdx.x] += red[threadIdx.x + off];
        __syncthreads();
    }
    if (threadIdx.x == 0) psums[b] = red[0];
}

__global__ void finalize_kernel(const float* __restrict__ psums,
                                float* __restrict__ out) {
    if (threadIdx.x == 0 && blockIdx.x == 0) {
        float t = 0.0f;
        #pragma unroll
        for (int b = 0; b < BATCH; ++b) t += psums[b];
        out[0] = t * (1.0f / (float)(BATCH * NPTS));
    }
}

extern "C" void kernel_launch(void* const* d_in, const int* in_sizes, int n_in,
                              void* d_out, int out_size, void* d_ws, size_t ws_size,
                              hipStream_t stream) {
    const float* pcs = (const float*)d_in[0];     // [B, N, 3] float32
    float* out = (float*)d_out;                   // scalar float32
    float* knn = (float*)d_ws;                    // [B, N] = 256 KB scratch
    float* psums = knn + (size_t)BATCH * NPTS;    // [B] per-batch penalty sums

    knn_tile_kernel<<<BATCH * NTILE, 32, 0, stream>>>(pcs, knn);
    batch_stats_kernel<<<BATCH, 256, 0, stream>>>(knn, psums);
    finalize_kernel<<<1, 32, 0, stream>>>(psums, out);
}